// SingleQubit_QuantumCircuit_89816356094198
// MI455X (gfx1250) — compile-verified
//
#include <hip/hip_runtime.h>
#include <hip/hip_bf16.h>

#define TPB  256
#define EPT  8
#define TILE (TPB * EPT)   // 2048 floats = 8 KB staged per block via TDM

typedef unsigned int u32x4 __attribute__((ext_vector_type(4)));
typedef int          i32x8 __attribute__((ext_vector_type(8)));
typedef int          i32x4 __attribute__((ext_vector_type(4)));
typedef float        f32x4 __attribute__((ext_vector_type(4)));

// Evaluate the 3-layer single-qubit circuit for one x, using pre-folded
// per-layer SU(2) matrices M_l = [[alpha, -conj(beta)], [beta, conj(alpha)]].
__device__ __forceinline__ float circuit_eval(float xv,
                                              const float* __restrict__ ar,
                                              const float* __restrict__ ai,
                                              const float* __restrict__ br,
                                              const float* __restrict__ bi) {
    float t = 0.5f * xv;
    float c = __cosf(t);
    float s = __sinf(t);
    // Layer 0: encoder Ry on (1,0) gives (c, s) (real); then apply M0.
    float are = ar[0] * c - br[0] * s;
    float aim = ai[0] * c + bi[0] * s;
    float bre = br[0] * c + ar[0] * s;
    float bim = bi[0] * c - ai[0] * s;
#pragma unroll
    for (int l = 1; l < 3; ++l) {
        // encoder Ry(x): real rotation on complex pair
        float nar = c * are - s * bre;
        float nai = c * aim - s * bim;
        float nbr = s * are + c * bre;
        float nbi = s * aim + c * bim;
        // M_l: a' = alpha*a - conj(beta)*b ; b' = beta*a + conj(alpha)*b
        are = ar[l] * nar - ai[l] * nai - br[l] * nbr - bi[l] * nbi;
        aim = ar[l] * nai + ai[l] * nar - br[l] * nbi + bi[l] * nbr;
        bre = br[l] * nar - bi[l] * nai + ar[l] * nbr + ai[l] * nbi;
        bim = br[l] * nai + bi[l] * nar + ar[l] * nbi - ai[l] * nbr;
    }
    return 0.5f * ((are * are + aim * aim) - (bre * bre + bim * bim) + 1.0f);
}

__global__ __launch_bounds__(TPB)
void qcirc_kernel(const float* __restrict__ x,
                  const float* __restrict__ th,
                  float* __restrict__ out,
                  int n) {
    extern __shared__ float tile[];  // TILE floats, filled by the TDM

    const long long base = (long long)blockIdx.x * TILE;

    // ---- Stage this block's input tile into LDS with the Tensor Data Mover.
    if (threadIdx.x < 32) {  // wave 0 issues once per block (EXEC ignored by TDM)
        unsigned long long ga  = (unsigned long long)(uintptr_t)(x + base);
        unsigned           lds = (unsigned)(uintptr_t)(&tile[0]);  // low 32b = LDS byte offset
        long long          remll = (long long)n - base;
        unsigned           rem   = (remll > TILE) ? (unsigned)TILE : (unsigned)remll;

        // D# group 0: count=1 | lds_addr | global_addr[56:0] | type=2
        u32x4 g0;
        g0[0] = 1u;
        g0[1] = lds;
        g0[2] = (unsigned)(ga & 0xFFFFFFFFu);
        g0[3] = (unsigned)((ga >> 32) & 0x01FFFFFFu) | (2u << 30);

        // D# group 1: data_size=4B; tensor_dim0=rem (OOB reads -> 0);
        // tensor_dim1=1; tile_dim0=TILE; tile_dim1=1; tile_dim2=0; stride0=TILE.
        i32x8 g1;
        g1[0] = (int)(2u << 16);                                   // data_size = 2 (4 bytes)
        g1[1] = (int)((rem & 0xFFFFu) << 16);                      // tensor_dim0[15:0]
        g1[2] = (int)(((rem >> 16) & 0xFFFFu) | (1u << 16));       // tensor_dim0[31:16], tensor_dim1 lo = 1
        g1[3] = (int)((unsigned)TILE << 16);                       // tensor_dim1 hi = 0, tile_dim0 = TILE
        g1[4] = 1;                                                 // tile_dim1 = 1, tile_dim2 = 0
        g1[5] = (int)TILE;                                         // tensor_dim0_stride lo
        g1[6] = 0;
        g1[7] = 0;

        i32x4 z4 = {0, 0, 0, 0};
#if __clang_major__ >= 23
        i32x8 z8 = {0, 0, 0, 0, 0, 0, 0, 0};
        __builtin_amdgcn_tensor_load_to_lds(g0, g1, z4, z4, z8, 0);
#else
        __builtin_amdgcn_tensor_load_to_lds(g0, g1, z4, z4, 0);
#endif
        __builtin_amdgcn_s_wait_tensorcnt(0);
        asm volatile("" ::: "memory");
    }

    // ---- Fold thetas into per-layer SU(2) matrices (uniform; overlaps TDM).
    // alpha_l = cos(hy) * e^{-i(hz0+hz1)},  beta_l = sin(hy) * e^{-i(hz0-hz1)}
    float ar[3], ai[3], br[3], bi[3];
#pragma unroll
    for (int l = 0; l < 3; ++l) {
        float hz0 = 0.5f * th[3 * l + 0];
        float hy  = 0.5f * th[3 * l + 1];
        float hz1 = 0.5f * th[3 * l + 2];
        float cy = __cosf(hy), sy = __sinf(hy);
        float p = hz0 + hz1, m = hz0 - hz1;
        ar[l] = cy * __cosf(p);
        ai[l] = -cy * __sinf(p);
        br[l] = sy * __cosf(m);
        bi[l] = -sy * __sinf(m);
    }

    __syncthreads();  // tile visible to all waves

    const int       tid = threadIdx.x;
    const long long i0  = base + (long long)tid * EPT;

    if (i0 + EPT <= (long long)n) {
        // Fast path: two b128 LDS reads, two coalesced b128 global stores (NT:
        // results are never re-read, keep them out of L2's retained set).
        const f32x4* tl = (const f32x4*)&tile[tid * EPT];
        f32x4 v0 = tl[0];
        f32x4 v1 = tl[1];
        f32x4 o0, o1;
        o0.x = circuit_eval(v0.x, ar, ai, br, bi);
        o0.y = circuit_eval(v0.y, ar, ai, br, bi);
        o0.z = circuit_eval(v0.z, ar, ai, br, bi);
        o0.w = circuit_eval(v0.w, ar, ai, br, bi);
        o1.x = circuit_eval(v1.x, ar, ai, br, bi);
        o1.y = circuit_eval(v1.y, ar, ai, br, bi);
        o1.z = circuit_eval(v1.z, ar, ai, br, bi);
        o1.w = circuit_eval(v1.w, ar, ai, br, bi);
        f32x4* op = (f32x4*)(out + i0);
        __builtin_nontemporal_store(o0, op);
        __builtin_nontemporal_store(o1, op + 1);
    } else {
        // Tail block: guarded scalar stores.
        for (int k = 0; k < EPT; ++k) {
            long long i = i0 + k;
            if (i < (long long)n) {
                float r = circuit_eval(tile[tid * EPT + k], ar, ai, br, bi);
                __builtin_nontemporal_store(r, out + i);
            }
        }
    }
}

extern "C" void kernel_launch(void* const* d_in, const int* in_sizes, int n_in,
                              void* d_out, int out_size, void* d_ws, size_t ws_size,
                              hipStream_t stream) {
    const float* x      = (const float*)d_in[0];   // (B,1) float32
    const float* thetas = (const float*)d_in[1];   // (3,3) float32
    float*       out    = (float*)d_out;           // (B,) float32
    int n = in_sizes[0];
    if (n <= 0) return;
    unsigned blocks = (unsigned)((n + TILE - 1) / TILE);
    qcirc_kernel<<<blocks, TPB, TILE * sizeof(float), stream>>>(x, thetas, out, n);
}